// MultiInputLSTMCell_44607530336618
// MI455X (gfx1250) — compile-verified
//
#include <hip/hip_runtime.h>
#include <math.h>

// ---------------------------------------------------------------------------
// MultiInputLSTMCell step, MI455X (gfx1250).
// Memory-bound problem (~134 MB of single-use f32 weights vs ~0.2 GFLOP), so:
//  - GEMVs (batch-1) -> split-K coalesced dot products w/ non-temporal loads.
//  - c_input @ AWh (16 x 2048 x 2048) -> fp32 WMMA 16x16x4 (M=16 == one tile).
//  - finalize kernel does the softmax-attention mix in fp32.
// ---------------------------------------------------------------------------

typedef __attribute__((ext_vector_type(2))) float v2f;
typedef __attribute__((ext_vector_type(8))) float v8f;

#define H    2048          // hidden size (== input size)
#define G3   6144          // 3*H gate width
#define W    16            // number of word cells

#define KS_GEMV 16         // split-K factor for GEMVs
#define KC_GEMV 128        // K chunk for GEMVs (16*128 = 2048)
#define KS_WMMA 8          // split-K factor for the WMMA GEMM
#define KC_WMMA 256        // K chunk for WMMA (8*256 = 2048)

// --------------------------------------------------------------------------
// Kernel 1: partial wih = input@Wi + h0@Wh   (columns of 3H, split-K)
// grid (G3/256, KS_GEMV), block 256. One thread per output column per chunk.
// Consecutive threads read consecutive addresses of Wi/Wh rows: coalesced.
// --------------------------------------------------------------------------
__global__ void gemv_wih(const float* __restrict__ x, const float* __restrict__ h,
                         const float* __restrict__ Wi, const float* __restrict__ Wh,
                         float* __restrict__ pwih) {
  __shared__ float sx[KC_GEMV];
  __shared__ float sh[KC_GEMV];
  const int tid = threadIdx.x;
  const int j   = blockIdx.x * 256 + tid;
  const int k0  = blockIdx.y * KC_GEMV;
  if (tid < KC_GEMV) { sx[tid] = x[k0 + tid]; sh[tid] = h[k0 + tid]; }
  __syncthreads();
  const float* wi = Wi + (size_t)k0 * G3 + j;
  const float* wh = Wh + (size_t)k0 * G3 + j;
  float acc = 0.f;
#pragma unroll 4
  for (int kk = 0; kk < KC_GEMV; ++kk) {
    acc += sx[kk] * __builtin_nontemporal_load(wi);   // weights are single-use
    acc += sh[kk] * __builtin_nontemporal_load(wh);
    wi += G3; wh += G3;
  }
  pwih[(size_t)blockIdx.y * G3 + j] = acc;
}

// --------------------------------------------------------------------------
// Kernel 2: partial alpha_wi = input@AWi   (columns of H, split-K)
// grid (H/256, KS_GEMV), block 256.
// --------------------------------------------------------------------------
__global__ void gemv_awi(const float* __restrict__ x, const float* __restrict__ AWi,
                         float* __restrict__ pawi) {
  __shared__ float sx[KC_GEMV];
  const int tid = threadIdx.x;
  const int j   = blockIdx.x * 256 + tid;
  const int k0  = blockIdx.y * KC_GEMV;
  if (tid < KC_GEMV) sx[tid] = x[k0 + tid];
  __syncthreads();
  const float* wp = AWi + (size_t)k0 * H + j;
  float acc = 0.f;
#pragma unroll 4
  for (int kk = 0; kk < KC_GEMV; ++kk) {
    acc += sx[kk] * __builtin_nontemporal_load(wp);
    wp += H;
  }
  pawi[(size_t)blockIdx.y * H + j] = acc;
}

// --------------------------------------------------------------------------
// Kernel 3: partial alpha_wh = c_input @ AWh via V_WMMA_F32_16X16X4_F32.
// c_input is (16, H): the entire M dimension is one WMMA tile.
// grid (H/16 n-tiles, KS_WMMA k-chunks), block 32 (one wave, EXEC all ones).
// Fragment layouts per ISA 7.12.2 (32-bit A 16x4 and C/D 16x16 tables):
//   A vgpr0 = K+0 (lanes 0-15) / K+2 (lanes 16-31), vgpr1 = K+1 / K+3
//   B mirrors A with rows striped across lanes (N = lane%16)
//   C vgpr r: lanes 0-15 -> row r, lanes 16-31 -> row r+8
// --------------------------------------------------------------------------
__global__ void gemm_awh_wmma(const float* __restrict__ Cw,    // c_input (16,H)
                              const float* __restrict__ AWh,   // (H,H)
                              float* __restrict__ pawh) {      // (KS_WMMA,16,H)
  const int lane = threadIdx.x;     // 0..31
  const int half = lane >> 4;       // 0 or 1
  const int m    = lane & 15;
  const int n0   = blockIdx.x * 16;
  const int kb   = blockIdx.y * KC_WMMA;

  const float* arow = Cw + (size_t)m * H;   // row m of c_input (L2-resident)
  v8f acc = {};

  for (int k = kb; k < kb + KC_WMMA; k += 4) {
    const int ka = k + 2 * half;
    v2f a, b;
    a.x = arow[ka];
    a.y = arow[ka + 1];
    b.x = __builtin_nontemporal_load(&AWh[(size_t)ka * H + n0 + m]);
    b.y = __builtin_nontemporal_load(&AWh[(size_t)(ka + 1) * H + n0 + m]);
    acc = __builtin_amdgcn_wmma_f32_16x16x4_f32(
        /*neg_a=*/false, a, /*neg_b=*/false, b,
        /*c_mod=*/(short)0, acc, /*reuse_a=*/false, /*reuse_b=*/false);
  }

  float* out = pawh + (size_t)blockIdx.y * (W * H);
#pragma unroll
  for (int r = 0; r < 8; ++r)
    out[(size_t)(r + 8 * half) * H + n0 + m] = acc[r];
}

// --------------------------------------------------------------------------
// Kernel 4: reduce partials + gates + attention softmax mix -> h1, c1.
// grid (H/256), block 256; one thread per hidden unit j.
// --------------------------------------------------------------------------
__global__ void finalize(const float* __restrict__ pwih,
                         const float* __restrict__ pawi,
                         const float* __restrict__ pawh,
                         const float* __restrict__ c_input,
                         const float* __restrict__ bWi, const float* __restrict__ bWh,
                         const float* __restrict__ b,
                         const float* __restrict__ bAWi, const float* __restrict__ ab,
                         const float* __restrict__ bAWh,
                         float* __restrict__ out) {
  const int j = blockIdx.x * blockDim.x + threadIdx.x;

  float wi_i = bWi[j]       + bWh[j]       + b[j];
  float wi_o = bWi[j + H]   + bWh[j + H]   + b[j + H];
  float wi_g = bWi[j + 2*H] + bWh[j + 2*H] + b[j + 2*H];
#pragma unroll
  for (int s = 0; s < KS_GEMV; ++s) {
    wi_i += pwih[(size_t)s * G3 + j];
    wi_o += pwih[(size_t)s * G3 + j + H];
    wi_g += pwih[(size_t)s * G3 + j + 2*H];
  }
  const float ig = 1.f / (1.f + expf(-wi_i));   // sigmoid(i)
  const float og = tanhf(wi_o);                 // tanh(o)
  const float gg = 1.f / (1.f + expf(-wi_g));   // sigmoid(g)

  float awi = bAWi[j] + ab[j];
#pragma unroll
  for (int s = 0; s < KS_GEMV; ++s) awi += pawi[(size_t)s * H + j];

  const float bh = bAWh[j];
  const float e0 = expf(ig);
  float denom = e0;
  float num   = gg * e0;
#pragma unroll
  for (int w = 0; w < W; ++w) {
    float awh = bh;
#pragma unroll
    for (int s = 0; s < KS_WMMA; ++s)
      awh += pawh[(size_t)s * (W * H) + (size_t)w * H + j];
    const float al = 1.f / (1.f + expf(-(awi + awh)));  // sigmoid
    const float e  = expf(al);
    denom += e;
    num   += c_input[(size_t)w * H + j] * e;
  }
  const float c1 = num / denom;
  const float h1 = og * tanhf(c1);
  out[j]     = h1;   // h1 first in tuple order
  out[H + j] = c1;
}

// --------------------------------------------------------------------------
extern "C" void kernel_launch(void* const* d_in, const int* in_sizes, int n_in,
                              void* d_out, int out_size, void* d_ws, size_t ws_size,
                              hipStream_t stream) {
  (void)in_sizes; (void)n_in; (void)out_size; (void)ws_size;
  const float* input   = (const float*)d_in[0];   // (1, 2048)
  const float* c_input = (const float*)d_in[1];   // (16, 2048)
  const float* h0      = (const float*)d_in[2];   // (1, 2048)
  // d_in[3] = c0 (unused by the reference)
  const float* Wi      = (const float*)d_in[4];   // (2048, 6144)
  const float* bWi     = (const float*)d_in[5];   // (6144)
  const float* Wh      = (const float*)d_in[6];   // (2048, 6144)
  const float* bWh     = (const float*)d_in[7];   // (6144)
  const float* b       = (const float*)d_in[8];   // (6144)
  const float* AWi     = (const float*)d_in[9];   // (2048, 2048)
  const float* bAWi    = (const float*)d_in[10];  // (2048)
  const float* AWh     = (const float*)d_in[11];  // (2048, 2048)
  const float* bAWh    = (const float*)d_in[12];  // (2048)
  const float* ab      = (const float*)d_in[13];  // (2048)
  float* out = (float*)d_out;                     // h1 (2048) ++ c1 (2048)

  // Workspace layout (floats): all partials written exactly once -> no init.
  float* pwih = (float*)d_ws;                  // KS_GEMV * G3   = 98304
  float* pawi = pwih + (size_t)KS_GEMV * G3;   // KS_GEMV * H    = 32768
  float* pawh = pawi + (size_t)KS_GEMV * H;    // KS_WMMA * W*H  = 262144
                                               // total 1.5 MB

  gemv_wih<<<dim3(G3 / 256, KS_GEMV), 256, 0, stream>>>(input, h0, Wi, Wh, pwih);
  gemv_awi<<<dim3(H / 256, KS_GEMV), 256, 0, stream>>>(input, AWi, pawi);
  gemm_awh_wmma<<<dim3(H / 16, KS_WMMA), 32, 0, stream>>>(c_input, AWh, pawh);
  finalize<<<dim3(H / 256), 256, 0, stream>>>(pwih, pawi, pawh, c_input,
                                              bWi, bWh, b, bAWi, ab, bAWh, out);
}